// RnnTest_11879879543264
// MI455X (gfx1250) — compile-verified
//
#include <hip/hip_runtime.h>
#include <hip/hip_bf16.h>

#define B_ 4096
#define T_ 2048

typedef float v2f __attribute__((ext_vector_type(2)));
typedef float v8f __attribute__((ext_vector_type(8)));

// ---------------- fast activations (CDNA5 v_tanh_f32 trans op) ----------------
__device__ __forceinline__ float fast_tanh(float v) {
#if __has_builtin(__builtin_amdgcn_tanhf)
    return __builtin_amdgcn_tanhf(v);
#else
    // tanh(v) = 1 - 2/(exp(2v)+1); exp(2v) = exp2(2*log2(e)*v)
    float e = __builtin_amdgcn_exp2f(v * 2.885390081777927f);
    return 1.0f - 2.0f * __builtin_amdgcn_rcpf(e + 1.0f);
#endif
}

__device__ __forceinline__ float fast_sigmoid(float v) {
    return fmaf(0.5f, fast_tanh(0.5f * v), 0.5f);
}

// ------------- quad broadcast via DPP quad_perm (no LDS on critical chain) ----
template <int K>
__device__ __forceinline__ float quad_bcast(float v) {
    int r = __builtin_amdgcn_update_dpp(
        0, __float_as_int(v),
        (K << 0) | (K << 2) | (K << 4) | (K << 6), /* quad_perm broadcast K */
        0xF, 0xF, true);
    return __int_as_float(r);
}

// ------------------------------- LSTM step -----------------------------------
// Lane = (batch b, hidden j); quad of lanes = one batch element.
__device__ __forceinline__ void lstm_step(float x0, float x1, float x2,
                                          const float wih[4][3],
                                          const float whh[4][4],
                                          float& h, float& c) {
    // gather all four h values of this quad (h is this lane's h_j)
    float h0 = quad_bcast<0>(h);
    float h1 = quad_bcast<1>(h);
    float h2 = quad_bcast<2>(h);
    float h3 = quad_bcast<3>(h);

    float g[4];
#pragma unroll
    for (int q = 0; q < 4; ++q) {   // q: 0=i 1=f 2=g 3=o ; gate row = 4q + j
        float a = wih[q][0] * x0;
        a = fmaf(wih[q][1], x1, a);
        a = fmaf(wih[q][2], x2, a);
        a = fmaf(whh[q][0], h0, a);
        a = fmaf(whh[q][1], h1, a);
        a = fmaf(whh[q][2], h2, a);
        a = fmaf(whh[q][3], h3, a);
        g[q] = a;
    }
    float ig = fast_sigmoid(g[0]);
    float fg = fast_sigmoid(g[1]);
    float gg = fast_tanh(g[2]);
    float og = fast_sigmoid(g[3]);
    c = fmaf(fg, c, ig * gg);
    h = og * fast_tanh(c);
}

// --------------------------- recurrence kernel -------------------------------
// 4 lanes per batch element; h,c,weights register-resident; x prefetched 4 steps
// ahead with 16B-aligned b128 loads.
__global__ void lstm_recur_kernel(const float* __restrict__ x,
                                  const float* __restrict__ w_ih,  // [16][3]
                                  const float* __restrict__ w_hh,  // [16][4]
                                  float* __restrict__ h_out) {     // [B][4]
    int gtid = blockIdx.x * blockDim.x + threadIdx.x;
    int b = gtid >> 2;
    int j = gtid & 3;

    // per-lane weight rows {j, 4+j, 8+j, 12+j}
    float wih[4][3];
    float whh[4][4];
#pragma unroll
    for (int q = 0; q < 4; ++q) {
        int r = 4 * q + j;
        wih[q][0] = w_ih[r * 3 + 0];
        wih[q][1] = w_ih[r * 3 + 1];
        wih[q][2] = w_ih[r * 3 + 2];
        whh[q][0] = w_hh[r * 4 + 0];
        whh[q][1] = w_hh[r * 4 + 1];
        whh[q][2] = w_hh[r * 4 + 2];
        whh[q][3] = w_hh[r * 4 + 3];
    }

    const float4* xp =
        reinterpret_cast<const float4*>(x + (size_t)b * (size_t)T_ * 3);

    float h = 0.0f, c = 0.0f;

    // chunk = 4 timesteps = 12 floats = 3 float4 (16B aligned)
    float4 c0 = xp[0], c1 = xp[1], c2 = xp[2];
    const int NCH = T_ / 4;
    for (int ch = 0; ch < NCH; ++ch) {
        float4 n0, n1, n2;
        if (ch + 1 < NCH) {   // prefetch next chunk; uniform branch per wave
            const float4* np = xp + 3 * (ch + 1);
            n0 = np[0];
            n1 = np[1];
            n2 = np[2];
        } else {
            n0 = c0; n1 = c1; n2 = c2;  // dummy, unused
        }
        lstm_step(c0.x, c0.y, c0.z, wih, whh, h, c);
        lstm_step(c0.w, c1.x, c1.y, wih, whh, h, c);
        lstm_step(c1.z, c1.w, c2.x, wih, whh, h, c);
        lstm_step(c2.y, c2.z, c2.w, wih, whh, h, c);
        c0 = n0; c1 = n1; c2 = n2;
    }

    h_out[b * 4 + j] = h;   // fully coalesced
}

// ------------------------- FC epilogue via WMMA ------------------------------
// out[4096] = H[4096x4] * fc_w[4] + fc_b, tiled as 16x16x4 f32 WMMA.
// A (16x4 f32, documented layout): VGPR0 = K0 (lanes 0-15) / K2 (lanes 16-31),
//                                  VGPR1 = K1 / K3, lane%16 = row M.
// B (4x16): fc_w placed in column 0 (mirrored layout).
// C: column 0 -> lane 0 holds rows 0-7 in VGPRs 0-7, lane 16 holds rows 8-15.
__global__ void lstm_fc_wmma_kernel(const float* __restrict__ hfin,  // [B][4]
                                    const float* __restrict__ fcw,   // [4]
                                    const float* __restrict__ fcb,   // [1]
                                    float* __restrict__ out) {       // [B]
    int gtid = blockIdx.x * blockDim.x + threadIdx.x;
    int wave = gtid >> 5;            // one 16-row tile per wave
    int lane = threadIdx.x & 31;
    int m = lane & 15;               // row within tile
    int hi = lane >> 4;              // 0: K0/K1   1: K2/K3
    int row = wave * 16 + m;

    // A-matrix: this lane supplies h[row][2*hi] and h[row][2*hi+1]
    v2f A;
    A.x = hfin[row * 4 + hi * 2 + 0];
    A.y = hfin[row * 4 + hi * 2 + 1];

    // B-matrix: fc_w down column 0 (branchless lane selects)
    float f0 = fcw[0], f1 = fcw[1], f2 = fcw[2], f3 = fcw[3];
    v2f Bm;
    Bm.x = (lane == 0) ? f0 : ((lane == 16) ? f2 : 0.0f);
    Bm.y = (lane == 0) ? f1 : ((lane == 16) ? f3 : 0.0f);

    v8f C = {0.0f, 0.0f, 0.0f, 0.0f, 0.0f, 0.0f, 0.0f, 0.0f};
    // (neg_a, A, neg_b, B, c_mod, C, reuse_a, reuse_b)
    C = __builtin_amdgcn_wmma_f32_16x16x4_f32(false, A, false, Bm, (short)0, C,
                                              false, false);

    float bias = fcb[0];
    if (m == 0) {   // lanes 0 and 16 hold column 0 of the result
        int base = wave * 16 + hi * 8;
#pragma unroll
        for (int r = 0; r < 8; ++r) out[base + r] = C[r] + bias;
    }
}

// ------------------------------- launcher ------------------------------------
extern "C" void kernel_launch(void* const* d_in, const int* in_sizes, int n_in,
                              void* d_out, int out_size, void* d_ws,
                              size_t ws_size, hipStream_t stream) {
    const float* x    = (const float*)d_in[0];  // [B,T,3]
    const float* w_ih = (const float*)d_in[1];  // [16,3]
    const float* w_hh = (const float*)d_in[2];  // [16,4]
    const float* fc_w = (const float*)d_in[3];  // [1,4]
    const float* fc_b = (const float*)d_in[4];  // [1]
    float* out  = (float*)d_out;                // [B]
    float* hbuf = (float*)d_ws;                 // B*4 floats scratch

    // recurrence: 4 lanes per batch element -> 16384 threads
    lstm_recur_kernel<<<dim3((B_ * 4) / 256), dim3(256), 0, stream>>>(
        x, w_ih, w_hh, hbuf);

    // FC epilogue: one wave per 16 rows -> 256 waves = 8192 threads
    lstm_fc_wmma_kernel<<<dim3((B_ / 16) * 32 / 256), dim3(256), 0, stream>>>(
        hbuf, fc_w, fc_b, out);
}